// StackedGCN_36893769073013
// MI455X (gfx1250) — compile-verified
//
#include <hip/hip_runtime.h>
#include <hip/hip_bf16.h>

#define N_NODES 50000
#define N_EDGES 800000
#define F_IN    128
#define HIDDEN  128
#define F_OUT   64
#define BN_EPS  1e-5f

typedef float v2f __attribute__((ext_vector_type(2)));
typedef float v8f __attribute__((ext_vector_type(8)));

// ---------------------------------------------------------------- zero fill
__global__ void __launch_bounds__(256) zero_kernel(float4* __restrict__ p, int n4) {
    int i = blockIdx.x * 256 + threadIdx.x;
    if (i < n4) p[i] = make_float4(0.f, 0.f, 0.f, 0.f);
}

// ---------------------------------------------------------------- SpMM (scatter-add)
// One wave (32 lanes) per edge; each lane owns a float4 slice of the 128-wide row.
__global__ void __launch_bounds__(256) spmm_kernel(
    const float* __restrict__ ev, const int* __restrict__ es,
    const int* __restrict__ ed, const float* __restrict__ x,
    float* __restrict__ agg)
{
    const int e    = blockIdx.x * 8 + (threadIdx.x >> 5);
    const int lane = threadIdx.x & 31;
    const int s = es[e];
    const int d = ed[e];
    const float w = ev[e];
    const float4 f = *(const float4*)(x + (size_t)s * 128 + lane * 4);
    float* p = agg + (size_t)d * 128 + lane * 4;
    unsafeAtomicAdd(p + 0, w * f.x);
    unsafeAtomicAdd(p + 1, w * f.y);
    unsafeAtomicAdd(p + 2, w * f.z);
    unsafeAtomicAdd(p + 3, w * f.w);
}

// ---------------------------------------------------------------- GEMM1 (+BN stats)
// h_pre = [agg | feat] @ W1 + b1 ; per-column sum & sumsq accumulated atomically.
// 16 rows x 128 cols per block; 8 waves, one 16x16 WMMA tile per wave; K = 256.
__global__ void __launch_bounds__(256) gemm1_kernel(
    const float* __restrict__ agg, const float* __restrict__ feat,
    const float* __restrict__ W1, const float* __restrict__ b1,
    float* __restrict__ h, float* __restrict__ stats)
{
    __shared__ float sA[16 * 260];   // 16 rows x 256 cols, stride 260 (bank-conflict free)
    const int t = threadIdx.x;
    const int rowBase = blockIdx.x * 16;

    // Stage concat tile [agg | feat] into LDS (1024 float4, 4 per thread)
    #pragma unroll
    for (int i = 0; i < 4; ++i) {
        int idx = t + 256 * i;          // float4 index
        int r   = idx >> 6;             // 64 float4 per row
        int col = (idx & 63) << 2;
        const float* srcp = (col < F_IN)
            ? (agg  + (size_t)(rowBase + r) * F_IN + col)
            : (feat + (size_t)(rowBase + r) * F_IN + (col - F_IN));
        *(float4*)(&sA[r * 260 + col]) = *(const float4*)srcp;
    }
    __syncthreads();

    const int wave  = t >> 5;            // column tile 0..7
    const int lane  = t & 31;
    const int m     = lane & 15;         // row (A) / col (B,C) within tile
    const int khalf = (lane >> 4) * 2;   // 0 or 2 (f32 16x4 fragment layout)
    const int n     = wave * 16 + m;     // output column 0..127

    v8f c = {};
    for (int k0 = 0; k0 < 2 * F_IN; k0 += 4) {
        v2f a, b;
        a.x = sA[m * 260 + k0 + khalf];
        a.y = sA[m * 260 + k0 + khalf + 1];
        b.x = W1[(size_t)(k0 + khalf)     * HIDDEN + n];
        b.y = W1[(size_t)(k0 + khalf + 1) * HIDDEN + n];
        c = __builtin_amdgcn_wmma_f32_16x16x4_f32(
                false, a, false, b, (short)0, c, false, false);
    }

    const float bias = b1[n];
    const int rOff = (lane >> 4) * 8;    // C layout: lanes 16-31 hold M=8..15
    float s1 = 0.f, s2 = 0.f;
    #pragma unroll
    for (int j = 0; j < 8; ++j) {
        float hv = c[j] + bias;
        h[(size_t)(rowBase + rOff + j) * HIDDEN + n] = hv;
        s1 += hv;
        s2 += hv * hv;
    }
    unsafeAtomicAdd(&stats[n], s1);
    unsafeAtomicAdd(&stats[HIDDEN + n], s2);
}

// ---------------------------------------------------------------- BN finalize
// stats[n] <- scale = gamma*invstd ; stats[128+n] <- shift = beta - mean*scale
__global__ void __launch_bounds__(128) bn_finalize_kernel(
    float* __restrict__ stats, const float* __restrict__ gamma,
    const float* __restrict__ beta)
{
    int n = threadIdx.x;
    float mean   = stats[n] * (1.0f / N_NODES);
    float var    = stats[HIDDEN + n] * (1.0f / N_NODES) - mean * mean;  // biased
    float invstd = rsqrtf(var + BN_EPS);
    float scale  = gamma[n] * invstd;
    stats[n]          = scale;
    stats[HIDDEN + n] = beta[n] - mean * scale;
}

// ---------------------------------------------------------------- BN + ReLU (in place)
__global__ void __launch_bounds__(256) bnrelu_kernel(
    float* __restrict__ h, const float* __restrict__ stats)
{
    int i   = blockIdx.x * 256 + threadIdx.x;  // float4 index, exact grid
    int col = (i & 31) * 4;                    // 32 float4 per 128-wide row
    float4 v = ((float4*)h)[i];
    v.x = fmaxf(0.f, v.x * stats[col + 0] + stats[HIDDEN + col + 0]);
    v.y = fmaxf(0.f, v.y * stats[col + 1] + stats[HIDDEN + col + 1]);
    v.z = fmaxf(0.f, v.z * stats[col + 2] + stats[HIDDEN + col + 2]);
    v.w = fmaxf(0.f, v.w * stats[col + 3] + stats[HIDDEN + col + 3]);
    ((float4*)h)[i] = v;
}

// ---------------------------------------------------------------- GEMM2
// out = [agg2 | h] @ W2 + b2 ; 16 rows x 64 cols per block; 4 waves.
__global__ void __launch_bounds__(128) gemm2_kernel(
    const float* __restrict__ agg, const float* __restrict__ h,
    const float* __restrict__ W2, const float* __restrict__ b2,
    float* __restrict__ out)
{
    __shared__ float sA[16 * 260];
    const int t = threadIdx.x;
    const int rowBase = blockIdx.x * 16;

    #pragma unroll
    for (int i = 0; i < 8; ++i) {
        int idx = t + 128 * i;
        int r   = idx >> 6;
        int col = (idx & 63) << 2;
        const float* srcp = (col < HIDDEN)
            ? (agg + (size_t)(rowBase + r) * HIDDEN + col)
            : (h   + (size_t)(rowBase + r) * HIDDEN + (col - HIDDEN));
        *(float4*)(&sA[r * 260 + col]) = *(const float4*)srcp;
    }
    __syncthreads();

    const int wave  = t >> 5;            // column tile 0..3
    const int lane  = t & 31;
    const int m     = lane & 15;
    const int khalf = (lane >> 4) * 2;
    const int n     = wave * 16 + m;     // output column 0..63

    v8f c = {};
    for (int k0 = 0; k0 < 2 * HIDDEN; k0 += 4) {
        v2f a, b;
        a.x = sA[m * 260 + k0 + khalf];
        a.y = sA[m * 260 + k0 + khalf + 1];
        b.x = W2[(size_t)(k0 + khalf)     * F_OUT + n];
        b.y = W2[(size_t)(k0 + khalf + 1) * F_OUT + n];
        c = __builtin_amdgcn_wmma_f32_16x16x4_f32(
                false, a, false, b, (short)0, c, false, false);
    }

    const float bias = b2[n];
    const int rOff = (lane >> 4) * 8;
    #pragma unroll
    for (int j = 0; j < 8; ++j) {
        out[(size_t)(rowBase + rOff + j) * F_OUT + n] = c[j] + bias;
    }
}

// ---------------------------------------------------------------- launch
extern "C" void kernel_launch(void* const* d_in, const int* in_sizes, int n_in,
                              void* d_out, int out_size, void* d_ws, size_t ws_size,
                              hipStream_t stream) {
    const float* features = (const float*)d_in[0];
    const float* W1       = (const float*)d_in[1];
    const float* b1       = (const float*)d_in[2];
    const float* gamma    = (const float*)d_in[3];
    const float* beta     = (const float*)d_in[4];
    const float* W2       = (const float*)d_in[5];
    const float* b2       = (const float*)d_in[6];
    const float* ev       = (const float*)d_in[7];
    const int*   es       = (const int*)d_in[8];
    const int*   ed       = (const int*)d_in[9];
    float* out = (float*)d_out;

    float* ws    = (float*)d_ws;
    float* agg   = ws;                                   // N_NODES*128 floats
    float* stats = ws + (size_t)N_NODES * 128;           // 256 floats
    float* h     = stats + 2 * HIDDEN;                   // N_NODES*128 floats

    const int aggStats4 = (N_NODES * 128 + 2 * HIDDEN) / 4;   // 1,600,064
    const int agg4      = (N_NODES * 128) / 4;                // 1,600,000

    // Layer 1
    zero_kernel<<<(aggStats4 + 255) / 256, 256, 0, stream>>>((float4*)agg, aggStats4);
    spmm_kernel<<<N_EDGES / 8, 256, 0, stream>>>(ev, es, ed, features, agg);
    gemm1_kernel<<<N_NODES / 16, 256, 0, stream>>>(agg, features, W1, b1, h, stats);
    bn_finalize_kernel<<<1, 128, 0, stream>>>(stats, gamma, beta);
    bnrelu_kernel<<<(N_NODES * 128 / 4) / 256, 256, 0, stream>>>(h, stats);

    // Layer 2 (reuse agg buffer)
    zero_kernel<<<(agg4 + 255) / 256, 256, 0, stream>>>((float4*)agg, agg4);
    spmm_kernel<<<N_EDGES / 8, 256, 0, stream>>>(ev, es, ed, h, agg);
    gemm2_kernel<<<N_NODES / 16, 128, 0, stream>>>(agg, h, W2, b2, out);
}